// FixedMoiraiGating_14516989460788
// MI455X (gfx1250) — compile-verified
//
#include <hip/hip_runtime.h>
#include <hip/hip_bf16.h>

typedef __attribute__((ext_vector_type(2))) float v2f;
typedef __attribute__((ext_vector_type(4))) float v4f;
typedef __attribute__((ext_vector_type(8))) float v8f;

#define D_DIM   4096          // reduction dim
#define E_DIM   64            // experts
#define KC      64            // K-chunk per pipeline stage
#define BM      32            // tokens per block
#define THREADS 256           // 8 wave32
#define XSTR    68            // LDS row stride (64 + 4 pad -> bank-conflict-free b64 frag loads)
#define NSTAGE  (D_DIM / KC)  // 64

__global__ __launch_bounds__(THREADS)
void moe_gate_topk_kernel(const float* __restrict__ x,
                          const float* __restrict__ W,
                          const float* __restrict__ bias,
                          float* __restrict__ out_probs,
                          int*   __restrict__ out_idx)
{
    __shared__ float xs[2][BM * XSTR];      // 17408 B  x-tile double buffer
    __shared__ float ws[2][E_DIM * XSTR];   // 34816 B  W-tile double buffer
    __shared__ float lg[BM * E_DIM];        //  8192 B  logits tile for top-k

    const int tid    = threadIdx.x;
    const int lane   = tid & 31;
    const int wave   = tid >> 5;
    const int wave_m = wave >> 2;           // 0..1  (token group of 16)
    const int wave_n = wave & 3;            // 0..3  (expert group of 16)
    const int row16  = lane & 15;           // M (A) / N-row (B) within the 16
    const int koff   = (lane >= 16) ? 2 : 0;// K offset per A/B f32 fragment layout

    const long tokBase = (long)blockIdx.x * BM;

    // ---- preload stage 0 into buffer 0 (coalesced b128) ----
    {
#pragma unroll
        for (int j = 0; j < 2; ++j) {               // 32 rows x 16 float4
            int item = tid + j * THREADS;
            int r = item >> 4, c4 = item & 15;
            v4f v = *(const v4f*)(x + (tokBase + r) * D_DIM + c4 * 4);
            *(v4f*)(&xs[0][r * XSTR + c4 * 4]) = v;
        }
#pragma unroll
        for (int j = 0; j < 4; ++j) {               // 64 rows x 16 float4
            int item = tid + j * THREADS;
            int r = item >> 4, c4 = item & 15;
            v4f v = *(const v4f*)(W + (long)r * D_DIM + c4 * 4);
            *(v4f*)(&ws[0][r * XSTR + c4 * 4]) = v;
        }
    }
    __syncthreads();

    v8f acc = {0.f,0.f,0.f,0.f,0.f,0.f,0.f,0.f};

    for (int s = 0; s < NSTAGE; ++s) {
        const int buf     = s & 1;
        const int nbuf    = buf ^ 1;
        const bool hasNext = (s + 1 < NSTAGE);
        const int k0n     = (s + 1) * KC;

        // 1) issue global loads for next chunk (latency hidden under WMMAs)
        v4f xv[2], wv[4];
        if (hasNext) {
#pragma unroll
            for (int j = 0; j < 2; ++j) {
                int item = tid + j * THREADS;
                int r = item >> 4, c4 = item & 15;
                xv[j] = *(const v4f*)(x + (tokBase + r) * D_DIM + k0n + c4 * 4);
            }
#pragma unroll
            for (int j = 0; j < 4; ++j) {
                int item = tid + j * THREADS;
                int r = item >> 4, c4 = item & 15;
                wv[j] = *(const v4f*)(W + (long)r * D_DIM + k0n + c4 * 4);
            }
        }

        // 2) compute current chunk: 16 x v_wmma_f32_16x16x4_f32
        const float* xb = &xs[buf][(wave_m * 16 + row16) * XSTR + koff];
        const float* wb = &ws[buf][(wave_n * 16 + row16) * XSTR + koff];
#pragma unroll
        for (int kk = 0; kk < KC; kk += 4) {
            v2f a  = *(const v2f*)(xb + kk);   // ds_load_b64, contiguous pair
            v2f bm = *(const v2f*)(wb + kk);
            acc = __builtin_amdgcn_wmma_f32_16x16x4_f32(
                      false, a, false, bm, (short)0, acc, false, false);
        }

        // 3) commit next chunk to the other LDS buffer
        if (hasNext) {
#pragma unroll
            for (int j = 0; j < 2; ++j) {
                int item = tid + j * THREADS;
                int r = item >> 4, c4 = item & 15;
                *(v4f*)(&xs[nbuf][r * XSTR + c4 * 4]) = xv[j];
            }
#pragma unroll
            for (int j = 0; j < 4; ++j) {
                int item = tid + j * THREADS;
                int r = item >> 4, c4 = item & 15;
                *(v4f*)(&ws[nbuf][r * XSTR + c4 * 4]) = wv[j];
            }
        }
        __syncthreads();
    }

    // ---- epilogue: bias add, dump C tile to LDS ----
    {
        const int expert = wave_n * 16 + row16;
        const float bv   = bias[expert];
        const int mAdd   = (lane >= 16) ? 8 : 0;   // C/D layout: lanes 16-31 hold M=r+8
#pragma unroll
        for (int r = 0; r < 8; ++r) {
            int tokLocal = wave_m * 16 + r + mAdd;
            lg[tokLocal * E_DIM + expert] = acc[r] + bv;
        }
    }
    __syncthreads();

    // ---- top-2 + softmax, one thread per token ----
    if (tid < BM) {
        const float* lrow = &lg[tid * E_DIM];
        float v0 = -__builtin_inff(), v1 = -__builtin_inff();
        int   i0 = 0, i1 = 0;
#pragma unroll 8
        for (int e = 0; e < E_DIM; ++e) {
            float v = lrow[e];
            if (v > v0)      { v1 = v0; i1 = i0; v0 = v; i0 = e; }
            else if (v > v1) { v1 = v;  i1 = e; }
        }
        // softmax over [v0, v1] (v0 >= v1, numerically stable)
        float e1  = __expf(v1 - v0);
        float inv = 1.0f / (1.0f + e1);
        long tok = tokBase + tid;
        out_probs[tok * 2 + 0] = inv;
        out_probs[tok * 2 + 1] = e1 * inv;
        out_idx[tok * 2 + 0]   = i0;
        out_idx[tok * 2 + 1]   = i1;
    }
}

extern "C" void kernel_launch(void* const* d_in, const int* in_sizes, int n_in,
                              void* d_out, int out_size, void* d_ws, size_t ws_size,
                              hipStream_t stream) {
    const float* x = (const float*)d_in[0];   // [B,S,D] = [2,4096,4096]
    const float* W = (const float*)d_in[1];   // [E,D]   = [64,4096]
    const float* b = (const float*)d_in[2];   // [E]

    const int n_tokens = in_sizes[0] / D_DIM; // 8192
    float* probs = (float*)d_out;                         // [B,S,2] f32
    int*   idx   = (int*)d_out + (size_t)n_tokens * 2;    // [B,S,2] i32 (bit-stored)

    dim3 grid(n_tokens / BM);
    moe_gate_topk_kernel<<<grid, THREADS, 0, stream>>>(x, W, b, probs, idx);
}